// DynamicalFeaturesExtractor_8005819040017
// MI455X (gfx1250) — compile-verified
//
#include <hip/hip_runtime.h>
#include <hip/hip_bf16.h>

// CSR edge record: one 8-byte load per edge in the hot loop.
struct __align__(8) EdgeRec {
    float w;
    int   s;
};

#define TPB   256            // 8 waves (wave32)
#define CHUNK 2048           // edges per LDS stage (16 KB)
#define IPT   (CHUNK / TPB)  // async b64 issues per thread per stage (= 8)

// ---------------------------------------------------------------------------
// CDNA5 async Global->LDS helpers (gfx1250; ASYNCcnt-tracked)
// ---------------------------------------------------------------------------
__device__ __forceinline__ void async_copy_b64(unsigned lds_byte_addr,
                                               const void* gaddr) {
    asm volatile("global_load_async_to_lds_b64 %0, %1, off"
                 :: "v"(lds_byte_addr), "v"(gaddr)
                 : "memory");
}
__device__ __forceinline__ void wait_asynccnt_le_ipt() {
    asm volatile("s_wait_asynccnt 0x8" ::: "memory");  // == IPT
}

__device__ __forceinline__ unsigned lds_offset_of(const void* p) {
    // Flat shared-aperture pointers keep the LDS byte offset in addr[31:0].
    return (unsigned)(unsigned long long)(uintptr_t)p;
}

// ---------------------------------------------------------------------------
// CSR-path kernels
// ---------------------------------------------------------------------------

__global__ void init_csr_kernel(const float* __restrict__ x,
                                const float* __restrict__ h_init,
                                float* __restrict__ h0,
                                int* __restrict__ cnt,
                                int n, int nin) {
    int i = blockIdx.x * blockDim.x + threadIdx.x;
    if (i < n) {
        h0[i]  = (i < nin) ? x[i] : h_init[i];
        cnt[i] = 0;
    }
}

__global__ void hist_kernel(const int* __restrict__ recv,
                            int* __restrict__ cnt, int e) {
    int i = blockIdx.x * blockDim.x + threadIdx.x;
    if (i < e) atomicAdd(&cnt[recv[i]], 1);
}

// Single-block exclusive prefix scan: cnt(counts) -> off[0..n]; cnt becomes cursor.
__global__ void scan_kernel(int* __restrict__ cnt,
                            int* __restrict__ off, int n) {
    __shared__ int part[1024];
    const int tid   = threadIdx.x;
    const int chunk = (n + 1023) >> 10;
    const int lo    = tid * chunk;
    const int hi    = (lo + chunk < n) ? (lo + chunk) : n;

    int s = 0;
    for (int k = lo; k < hi; ++k) s += cnt[k];
    part[tid] = s;
    __syncthreads();
    for (int d = 1; d < 1024; d <<= 1) {
        int v = (tid >= d) ? part[tid - d] : 0;
        __syncthreads();
        part[tid] += v;
        __syncthreads();
    }
    int run = part[tid] - s;
    for (int k = lo; k < hi; ++k) {
        int d  = cnt[k];
        off[k] = run;
        cnt[k] = run;
        run += d;
    }
    if (tid == 1023) off[n] = part[1023];
}

__global__ void scatter_kernel(const float* __restrict__ ew,
                               const int* __restrict__ snd,
                               const int* __restrict__ recv,
                               int* __restrict__ cursor,
                               EdgeRec* __restrict__ edges, int e) {
    int i = blockIdx.x * blockDim.x + threadIdx.x;
    if (i < e) {
        int r = recv[i];
        int p = atomicAdd(&cursor[r], 1);
        EdgeRec er;
        er.w = ew[i];
        er.s = snd[i];
        edges[p] = er;
    }
}

// ---------------------------------------------------------------------------
// Fused iteration with double-buffered async Global->LDS edge staging.
// Block owns nodes [i0, i0+TPB): their CSR slice is contiguous in `edges`.
// Accumulation is wave-cooperative (32 lanes x consecutive LDS reads ->
// conflict-free ds_load_b64) with shfl reduction.
// ---------------------------------------------------------------------------
__global__ void step_csr_async_kernel(const float* __restrict__ h,
                                      float* __restrict__ hn,
                                      float* __restrict__ out,
                                      const int* __restrict__ off,
                                      const EdgeRec* __restrict__ edges,
                                      int n, int t, int iters, int e_total) {
    __shared__ EdgeRec sbuf[2][CHUNK];
    __shared__ float   accbuf[TPB];

    const int tid  = threadIdx.x;
    const int lane = tid & 31;
    const int wave = tid >> 5;
    const int i0   = blockIdx.x * TPB;
    const int my   = i0 + tid;

    accbuf[tid] = 0.0f;

    // Per-thread CSR bounds (also broadcast within the wave later).
    int a_t = 0, b_t = 0;
    if (my < n) { a_t = off[my]; b_t = off[my + 1]; }

    if (my < n) out[(long)my * iters + t] = h[my];

    const int iend  = (i0 + TPB < n) ? (i0 + TPB) : n;
    const int blk_a = off[i0];
    const int blk_b = off[iend];
    const int nch   = (blk_b - blk_a + CHUNK - 1) / CHUNK;

    // Issue chunk 0 (every thread issues exactly IPT b64 copies, clamped).
    if (nch > 0) {
        for (int j = 0; j < IPT; ++j) {
            int idx = tid + j * TPB;
            int g   = blk_a + idx;
            g = (g < e_total - 1) ? g : (e_total - 1);
            async_copy_b64(lds_offset_of(&sbuf[0][idx]), &edges[g]);
        }
    }

    for (int c = 0; c < nch; ++c) {
        const int c0 = blk_a + c * CHUNK;
        const int c1 = (c0 + CHUNK < blk_b) ? (c0 + CHUNK) : blk_b;

        // Prefetch next chunk into the other buffer (always IPT issues so the
        // constant-immediate asynccnt wait below is exact; clamped in-bounds).
        {
            const int nb = (c + 1) & 1;
            const int nbase = c0 + CHUNK;
            for (int j = 0; j < IPT; ++j) {
                int idx = tid + j * TPB;
                int g   = nbase + idx;
                g = (g < e_total - 1) ? g : (e_total - 1);
                async_copy_b64(lds_offset_of(&sbuf[nb][idx]), &edges[g]);
            }
        }

        wait_asynccnt_le_ipt();   // chunk c's copies (issued earlier) are done
        __syncthreads();          // ... for all waves in the block

        const EdgeRec* buf = sbuf[c & 1];

        // Wave-cooperative accumulate: wave handles nodes [wave*32, wave*32+32).
        for (int jn = 0; jn < 32; ++jn) {
            const int node = i0 + wave * 32 + jn;
            if (node >= n) break;
            const int a = __shfl(a_t, jn, 32);
            const int b = __shfl(b_t, jn, 32);
            const int lo = (a > c0) ? a : c0;
            const int hi = (b < c1) ? b : c1;
            if (lo >= hi) continue;
            float p = 0.0f;
            for (int k = lo + lane; k < hi; k += 32) {
                EdgeRec ed = buf[k - c0];           // conflict-free ds_load_b64
                p += ed.w * h[ed.s];                // scattered L2 gather
            }
            for (int o = 16; o; o >>= 1) p += __shfl_down(p, o, 32);
            if (lane == 0) accbuf[wave * 32 + jn] += p;
        }
        __syncthreads();          // everyone done reading buf before reuse
    }

    if (my < n) hn[my] = tanhf(accbuf[tid]);
    // Leftover prefetch asyncs drain at S_ENDPGM (implicit wait-idle).
}

// ---------------------------------------------------------------------------
// Atomic fallback (small scratch)
// ---------------------------------------------------------------------------
__global__ void init_atomic_kernel(const float* __restrict__ x,
                                   const float* __restrict__ h_init,
                                   float* __restrict__ h0,
                                   float* __restrict__ acc,
                                   int n, int nin) {
    int i = blockIdx.x * blockDim.x + threadIdx.x;
    if (i < n) {
        h0[i]  = (i < nin) ? x[i] : h_init[i];
        acc[i] = 0.0f;
    }
}

__global__ void edge_atomic_kernel(const float* __restrict__ ew,
                                   const int* __restrict__ snd,
                                   const int* __restrict__ recv,
                                   const float* __restrict__ h,
                                   float* __restrict__ acc, int e) {
    int i = blockIdx.x * blockDim.x + threadIdx.x;
    if (i < e) atomicAdd(&acc[recv[i]], ew[i] * h[snd[i]]);
}

__global__ void node_finish_kernel(const float* __restrict__ h,
                                   float* __restrict__ hn,
                                   float* __restrict__ acc,
                                   float* __restrict__ out,
                                   int n, int t, int iters) {
    int i = blockIdx.x * blockDim.x + threadIdx.x;
    if (i >= n) return;
    out[(long)i * iters + t] = h[i];
    hn[i]  = tanhf(acc[i]);
    acc[i] = 0.0f;
}

// ---------------------------------------------------------------------------

static inline size_t align_up_256(size_t v) { return (v + 255) & ~(size_t)255; }

extern "C" void kernel_launch(void* const* d_in, const int* in_sizes, int n_in,
                              void* d_out, int out_size, void* d_ws, size_t ws_size,
                              hipStream_t stream) {
    const float* edge_w    = (const float*)d_in[0];
    const float* x         = (const float*)d_in[1];
    const float* h_init    = (const float*)d_in[2];
    const int*   senders   = (const int*)d_in[3];
    const int*   receivers = (const int*)d_in[4];

    const int e     = in_sizes[0];
    const int nin   = in_sizes[1];
    const int n     = in_sizes[2];
    const int iters = (n > 0) ? (out_size / n) : 0;
    float* out = (float*)d_out;

    const int bn = (n + TPB - 1) / TPB;
    const int be = (e + TPB - 1) / TPB;

    const size_t sz_edges = align_up_256((size_t)e * sizeof(EdgeRec));
    const size_t sz_h     = align_up_256((size_t)n * sizeof(float));
    const size_t sz_off   = align_up_256((size_t)(n + 1) * sizeof(int));
    const size_t sz_cnt   = align_up_256((size_t)n * sizeof(int));
    const size_t need_csr = sz_edges + 2 * sz_h + sz_off + sz_cnt;

    char* base = (char*)d_ws;

    if (ws_size >= need_csr) {
        EdgeRec* edges = (EdgeRec*)base;
        float*   h_a   = (float*)(base + sz_edges);
        float*   h_b   = (float*)(base + sz_edges + sz_h);
        int*     off   = (int*)(base + sz_edges + 2 * sz_h);
        int*     cnt   = (int*)(base + sz_edges + 2 * sz_h + sz_off);

        init_csr_kernel<<<bn, TPB, 0, stream>>>(x, h_init, h_a, cnt, n, nin);
        hist_kernel<<<be, TPB, 0, stream>>>(receivers, cnt, e);
        scan_kernel<<<1, 1024, 0, stream>>>(cnt, off, n);
        scatter_kernel<<<be, TPB, 0, stream>>>(edge_w, senders, receivers, cnt, edges, e);

        float* cur = h_a;
        float* nxt = h_b;
        for (int t = 0; t < iters; ++t) {
            step_csr_async_kernel<<<bn, TPB, 0, stream>>>(cur, nxt, out, off,
                                                          edges, n, t, iters, e);
            float* tmp = cur; cur = nxt; nxt = tmp;
        }
    } else {
        float* h_a = (float*)base;
        float* h_b = (float*)(base + sz_h);
        float* acc = (float*)(base + 2 * sz_h);

        init_atomic_kernel<<<bn, TPB, 0, stream>>>(x, h_init, h_a, acc, n, nin);

        float* cur = h_a;
        float* nxt = h_b;
        for (int t = 0; t < iters; ++t) {
            edge_atomic_kernel<<<be, TPB, 0, stream>>>(edge_w, senders, receivers, cur, acc, e);
            node_finish_kernel<<<bn, TPB, 0, stream>>>(cur, nxt, acc, out, n, t, iters);
            float* tmp = cur; cur = nxt; nxt = tmp;
        }
    }
}